// SinkhornDistanceGivenC_17781164605779
// MI455X (gfx1250) — compile-verified
//
#include <hip/hip_runtime.h>

// ---------------------------------------------------------------------------
// Sinkhorn (given C) on MI455X / gfx1250.
//
// Math: with K = exp(-C/eps), a = exp(v/eps), b = exp(u/eps):
//   u_new_i = eps*log_mu - eps*log( (K  a)_i )   (u cancels out of the LSE)
//   v_new_j = eps*log_nu - eps*log( (K' b)_j )
//   b_i     = (1/N + 1e-8) / rowsum_i            (no exp needed)
//   a_j     = (1/N + 1e-8) / colsum_j
//   pi      = K_ij * b_i * a_j ,  cost_b = sum(pi * C)
//
// K (64 MB) is L2-resident on MI455X (192 MB L2) -> the 100-iteration loop is
// two L2-bandwidth-bound GEMV passes per iteration instead of ~33M exp()s.
// Column pass uses V_WMMA_F32_16X16X4_F32 for the MAC accumulation.
// ---------------------------------------------------------------------------

typedef __attribute__((ext_vector_type(2))) float v2f;
typedef __attribute__((ext_vector_type(8))) float v8f;

#define BATCH 16
#define NDIM  1024
#define EPSV  0.1f
#define MU_P  (1.0f / 1024.0f + 1e-8f)   // exp(log_mu) = 1/N + 1e-8
#define THRESH 0.1f
#define NITER  100

// Static device scratch (capture-safe: no hipMalloc; re-initialized each call).
__device__ float g_K[(size_t)BATCH * NDIM * NDIM];  // exp(-C/eps), 64 MB, L2-resident
__device__ float g_u[BATCH * NDIM];                 // u (for err only)
__device__ float g_bexp[BATCH * NDIM];              // exp(u/eps)
__device__ float g_aexp[BATCH * NDIM];              // exp(v/eps)
__device__ float g_errpart[2048];                   // row-pass |du| partials
__device__ float g_costpart[4096];                  // finalize cost partials
__device__ int   g_done[NITER + 1];                 // done flag per iteration

// ---------------------------------------------------------------------------
__global__ __launch_bounds__(256) void k_init() {
  for (int i = threadIdx.x; i < BATCH * NDIM; i += 256) {
    g_u[i] = 0.0f;     // u0 = 0
    g_aexp[i] = 1.0f;  // v0 = 0 -> exp(v/eps) = 1
  }
  if (threadIdx.x == 0) g_done[0] = 0;
}

// K = exp(-C/eps); also emit the C copy into d_out's third segment.
__global__ __launch_bounds__(256) void k_precompute(const float* __restrict__ C,
                                                    float* __restrict__ outC) {
  size_t base = ((size_t)blockIdx.x * 256 + threadIdx.x) * 16;
  const float4* c4 = (const float4*)(C + base);
  float4* k4 = (float4*)(g_K + base);
  float4* o4 = (float4*)(outC + base);
#pragma unroll
  for (int s = 0; s < 4; ++s) {
    float4 c = c4[s];
    float4 k;
    k.x = expf(-10.0f * c.x);
    k.y = expf(-10.0f * c.y);
    k.z = expf(-10.0f * c.z);
    k.w = expf(-10.0f * c.w);
    k4[s] = k;
    o4[s] = c;
  }
}

// Row pass: rowsum_i = sum_j K[b,i,j]*aexp[b,j]; update u, bexp, err partials.
// Grid: 16 batches * 128 row-groups = 2048 blocks, 256 thr (wave per row).
__global__ __launch_bounds__(256) void k_row(int t) {
  __shared__ float sa[NDIM];
  __shared__ float sdu[8];
  if (g_done[t]) return;  // frozen: u,v unchanged
  int b = blockIdx.x >> 7;
  int rg = blockIdx.x & 127;
  int wave = threadIdx.x >> 5, lane = threadIdx.x & 31;
  ((float4*)sa)[threadIdx.x] = ((const float4*)(g_aexp + b * NDIM))[threadIdx.x];
  __syncthreads();

  int row = rg * 8 + wave;
  const float4* rowp =
      (const float4*)(g_K + ((size_t)b << 20) + (size_t)row * NDIM);
  const float4* sa4 = (const float4*)sa;
  float acc = 0.0f;
#pragma unroll
  for (int s = 0; s < 8; ++s) {                 // 512B contiguous per wave step
    float4 kv = rowp[lane + s * 32];
    float4 av = sa4[lane + s * 32];
    acc += kv.x * av.x + kv.y * av.y + kv.z * av.z + kv.w * av.w;
  }
  for (int off = 16; off > 0; off >>= 1) acc += __shfl_xor(acc, off, 32);

  if (lane == 0) {
    float rowsum = acc;
    float unew = EPSV * (logf(MU_P) - logf(rowsum));
    int gi = b * NDIM + row;
    sdu[wave] = fabsf(unew - g_u[gi]);
    g_u[gi] = unew;
    g_bexp[gi] = MU_P / rowsum;                 // exp(u_new/eps), no expf needed
  }
  __syncthreads();
  if (threadIdx.x == 0) {
    float s = 0.0f;
#pragma unroll
    for (int w = 0; w < 8; ++w) s += sdu[w];    // fixed order -> deterministic
    g_errpart[blockIdx.x] = s;
  }
}

// Column pass via WMMA: colsum_j = sum_i K[b,i,j]*bexp[b,i].
// Wave handles a 16-column tile over a 512-row half; per K-step of 4 rows:
//   B(4x16) = K tile (j across lanes 0..15 -> coalesced), A(16x4) = bexp
//   broadcast across M. D accumulates column sums replicated over all rows.
// A and B use identical k-slot conventions, so any HW k-permutation cancels.
// Block 0 additionally reduces the row-pass err partials and writes done[t+1]
// (everyone reads done[t]; only done[t+1] is written -> no race).
// Grid: 16 batches * 16 col-groups = 256 blocks, 256 thr (8 waves).
__global__ __launch_bounds__(256) void k_col(int t) {
  __shared__ float sb[NDIM];
  __shared__ float part[8][16];
  __shared__ float sred[256];
  bool isdone = (g_done[t] != 0);
  if (isdone) {
    if (blockIdx.x == 0 && threadIdx.x == 0) g_done[t + 1] = 1;
    return;
  }
  int b = blockIdx.x >> 4;
  int jg = blockIdx.x & 15;                 // 64 columns per block
  int wave = threadIdx.x >> 5, lane = threadIdx.x & 31;
  int tile = wave >> 1;                     // 4 tiles of 16 columns
  int ihalf = wave & 1;                     // rows [0,512) vs [512,1024)
  ((float4*)sb)[threadIdx.x] = ((const float4*)(g_bexp + b * NDIM))[threadIdx.x];
  __syncthreads();

  int half = lane >> 4;                     // K-slot half within the wave
  int j0 = jg * 64 + tile * 16;
  const float* Kc = g_K + ((size_t)b << 20) +
                    (size_t)(ihalf * 512 + 2 * half) * NDIM + j0 + (lane & 15);
  const float* sbh = sb + ihalf * 512 + 2 * half;

  v8f acc = {};
#pragma unroll 4
  for (int i = 0; i < 512; i += 4) {
    v2f A, Bm;
    A[0] = sbh[i];                          // bexp rows i / i+2 (per half)
    A[1] = sbh[i + 1];                      // bexp rows i+1 / i+3
    Bm[0] = Kc[(size_t)i * NDIM];           // K rows i / i+2, col j0+lane
    Bm[1] = Kc[(size_t)i * NDIM + NDIM];    // K rows i+1 / i+3
    acc = __builtin_amdgcn_wmma_f32_16x16x4_f32(
        false, A, false, Bm, (short)0, acc, false, false);
  }
  // D is row-replicated -> every component equals the column sum for this lane.
  float partial = acc[0];
  if (lane < 16) part[wave][lane] = partial;
  __syncthreads();
  if (threadIdx.x < 64) {                   // combine the two row-halves
    int tt = threadIdx.x >> 4, l16 = threadIdx.x & 15;
    float colsum = part[2 * tt][l16] + part[2 * tt + 1][l16];
    int j = jg * 64 + tt * 16 + l16;
    g_aexp[b * NDIM + j] = MU_P / colsum;   // exp(v_new/eps), no expf needed
  }

  if (blockIdx.x == 0) {                    // err reduction + done update
    float s = 0.0f;
#pragma unroll
    for (int k = 0; k < 8; ++k) s += g_errpart[threadIdx.x + k * 256];
    sred[threadIdx.x] = s;
    __syncthreads();
    for (int off = 128; off > 0; off >>= 1) {
      if (threadIdx.x < off) sred[threadIdx.x] += sred[threadIdx.x + off];
      __syncthreads();
    }
    if (threadIdx.x == 0)
      g_done[t + 1] = (sred[0] * (1.0f / 16.0f) < THRESH) ? 1 : 0;
  }
}

// pi = K * bexp_i * aexp_j -> d_out; cost partials per block.
// Grid: 16 batches * 256 chunks = 4096 blocks; 16 elems/thread (one row each).
__global__ __launch_bounds__(256) void k_final(const float* __restrict__ C,
                                               float* __restrict__ out) {
  __shared__ float sred[256];
  int b = blockIdx.x >> 8;
  int chunk = blockIdx.x & 255;
  size_t inbatch = (size_t)chunk * 4096 + (size_t)threadIdx.x * 16;
  size_t base = ((size_t)b << 20) + inbatch;
  int row = (int)(inbatch >> 10);
  int jb = (int)(inbatch & 1023);
  float bex = g_bexp[b * NDIM + row];
  const float4* k4 = (const float4*)(g_K + base);
  const float4* c4 = (const float4*)(C + base);
  const float4* a4 = (const float4*)(g_aexp + b * NDIM + jb);
  float4* p4 = (float4*)(out + 16 + base);          // pi segment
  float costacc = 0.0f;
#pragma unroll
  for (int s = 0; s < 4; ++s) {
    float4 kv = k4[s], cv = c4[s], av = a4[s];
    float4 pi;
    pi.x = kv.x * bex * av.x;
    pi.y = kv.y * bex * av.y;
    pi.z = kv.z * bex * av.z;
    pi.w = kv.w * bex * av.w;
    p4[s] = pi;
    costacc += pi.x * cv.x + pi.y * cv.y + pi.z * cv.z + pi.w * cv.w;
  }
  sred[threadIdx.x] = costacc;
  __syncthreads();
  for (int off = 128; off > 0; off >>= 1) {
    if (threadIdx.x < off) sred[threadIdx.x] += sred[threadIdx.x + off];
    __syncthreads();
  }
  if (threadIdx.x == 0) g_costpart[blockIdx.x] = sred[0];
}

// cost[b] = deterministic reduction of the 256 per-block partials.
__global__ __launch_bounds__(256) void k_cost(float* __restrict__ out) {
  __shared__ float sred[256];
  sred[threadIdx.x] = g_costpart[blockIdx.x * 256 + threadIdx.x];
  __syncthreads();
  for (int off = 128; off > 0; off >>= 1) {
    if (threadIdx.x < off) sred[threadIdx.x] += sred[threadIdx.x + off];
    __syncthreads();
  }
  if (threadIdx.x == 0) out[blockIdx.x] = sred[0];
}

// ---------------------------------------------------------------------------
extern "C" void kernel_launch(void* const* d_in, const int* in_sizes, int n_in,
                              void* d_out, int out_size, void* d_ws,
                              size_t ws_size, hipStream_t stream) {
  (void)in_sizes; (void)n_in; (void)out_size; (void)d_ws; (void)ws_size;
  const float* C = (const float*)d_in[2];   // x (d_in[0]), y (d_in[1]) unused
  float* out = (float*)d_out;               // [cost:16][pi:16M][C:16M]

  k_init<<<1, 256, 0, stream>>>();
  k_precompute<<<4096, 256, 0, stream>>>(C, out + 16 + (size_t)BATCH * NDIM * NDIM);
  for (int t = 0; t < NITER; ++t) {
    k_row<<<2048, 256, 0, stream>>>(t);
    k_col<<<256, 256, 0, stream>>>(t);
  }
  k_final<<<4096, 256, 0, stream>>>(C, out);
  k_cost<<<BATCH, 256, 0, stream>>>(out);
}